// MOERecLayer_84155589198302
// MI455X (gfx1250) — compile-verified
//
#include <hip/hip_runtime.h>
#include <math.h>

// Problem constants (from reference)
#define U_   20000
#define I_   10000
#define N_   32
#define D_   128
#define CCD_ 128
#define KC_  32
#define KS_  8
#define RST_BASE ((size_t)U_ * KC_ * D_)   // 81,920,000 floats

typedef __attribute__((ext_vector_type(2))) float v2f;
typedef __attribute__((ext_vector_type(8))) float v8f;

static __device__ __forceinline__ v8f wmma_f32(v2f a, v2f b, v8f c) {
  // D = A(16x4,f32) * B(4x16,f32) + C(16x16,f32)  -> v_wmma_f32_16x16x4_f32
  return __builtin_amdgcn_wmma_f32_16x16x4_f32(false, a, false, b, (short)0, c,
                                               false, false);
}

// ---------------- out-degree normalization prep ----------------
__global__ void deg_zero_kernel(float* deg) {
  int i = blockIdx.x * 256 + threadIdx.x;
  if (i < I_) deg[i] = 0.0f;
}

__global__ void deg_count_kernel(const int* __restrict__ nbrs, float* deg) {
  int i = blockIdx.x * 256 + threadIdx.x;
  if (i < U_ * N_) atomicAdd(&deg[nbrs[i]], 1.0f);
}

__global__ void deg_norm_kernel(float* deg) {
  int i = blockIdx.x * 256 + threadIdx.x;
  if (i < I_) deg[i] = rsqrtf(fmaxf(deg[i], 1.0f));
}

// ---------------- main kernel: 1 wave32 handles TWO users ----------------
#define MS 132   // mailbox/sel/proj LDS row stride: conflict-free v2f frags
#define GS 34    // 32x32 sims matrix stride
#define LS 18    // logits/weights row stride (even -> 8B-aligned v2f)

// Gram + cdist + sims + greedy selection for one user's 32x128 mailbox.
__device__ __forceinline__ void gram_greedy(const float* __restrict__ mrow,
                                            float* __restrict__ smat,
                                            const int L, const int hl,
                                            const int hs, int* sels) {
  // Gram G = M * M^T (32x32, K=128) via V_WMMA_F32_16X16X4_F32
  {
    v8f g00 = {}, g01 = {}, g10 = {}, g11 = {};
    for (int k0 = 0; k0 < D_; k0 += 4) {
      const int kk = k0 + 2 * hs;
      const v2f a0 = *(const v2f*)&mrow[( 0 + hl) * MS + kk];
      const v2f a1 = *(const v2f*)&mrow[(16 + hl) * MS + kk];
      g00 = wmma_f32(a0, a0, g00);
      g01 = wmma_f32(a0, a1, g01);
      g10 = wmma_f32(a1, a0, g10);
      g11 = wmma_f32(a1, a1, g11);
    }
#pragma unroll
    for (int v = 0; v < 8; ++v) {
      const int r = v + 8 * hs;
      smat[( 0 + r) * GS +  0 + hl] = g00[v];
      smat[( 0 + r) * GS + 16 + hl] = g01[v];
      smat[(16 + r) * GS +  0 + hl] = g10[v];
      smat[(16 + r) * GS + 16 + hl] = g11[v];
    }
  }
  __syncthreads();

  // cdist + mean; lane L owns row n=L (diag read happens before overwrite)
  float rowsum = 0.0f;
  {
    const float sqn = smat[L * GS + L];
    for (int m = 0; m < N_; ++m) {
      const float sqm = smat[m * GS + m];
      const float g   = smat[L * GS + m];
      const float d   = sqrtf(fmaxf(sqn + sqm - 2.0f * g, 0.0f));
      smat[L * GS + m] = d;
      rowsum += d;
    }
  }
  for (int o = 16; o; o >>= 1) rowsum += __shfl_xor(rowsum, o, 32);
  const float inv_mean = 1024.0f / rowsum;       // 1/(SIGMA*mean), SIGMA=1
  for (int m = 0; m < N_; ++m)
    smat[L * GS + m] = expf(-smat[L * GS + m] * inv_mean);
  __syncthreads();

  // greedy facility-location selection (K_SELECT=8)
  float cache = 0.0f;  // lane L owns cache[m=L]
#pragma unroll
  for (int it = 0; it < KS_; ++it) {
    float gain = 0.0f;  // lane L computes gain for row n=L
    for (int m = 0; m < N_; ++m) {
      const float cm = __shfl(cache, m, 32);
      const float s  = smat[L * GS + m];
      gain += fmaxf(s, cm) - cm;
    }
    int best = L; float bg = gain;                 // argmax, low-index ties
    for (int o = 16; o; o >>= 1) {
      const float og = __shfl_xor(bg, o, 32);
      const int   ob = __shfl_xor(best, o, 32);
      if (og > bg || (og == bg && ob < best)) { bg = og; best = ob; }
    }
    sels[it] = best;
    cache = fmaxf(cache, smat[best * GS + L]);
  }
  __syncthreads();  // smat/mail may be reused for the next user
}

__launch_bounds__(32)
__global__ void moerec_kernel(const float* __restrict__ h_item,
                              const int*   __restrict__ nbrs,
                              const float* __restrict__ Wp,      // [D, CCD]
                              const float* __restrict__ codes,   // [KC, CCD]
                              const float* __restrict__ nrm_src, // [I]
                              float*       __restrict__ out) {
  __shared__ float mail[N_ * MS];   // ONE 32x128 mailbox, reused per user
  __shared__ float sel [16 * MS];   // rows 0-7: user0 sel, 8-15: user1 sel
  __shared__ float pj  [16 * MS];   // tanh(S*W), packed same way
  __shared__ float smat[N_ * GS];   // Gram -> dist -> sims (per user)
  __shared__ float lg  [KC_ * LS];  // logits[code][k packed 0..15]
  __shared__ float wt  [KC_ * LS];  // softmax weights, same packing

  const int u0 = blockIdx.x * 2;
  const int L  = threadIdx.x;  // 0..31
  const int hl = L & 15;
  const int hs = L >> 4;

  // ---- 1-5. per user: gather -> Gram/sims/greedy -> pack selected rows ----
  for (int usr = 0; usr < 2; ++usr) {
    // gather mailbox: mail[n][d] = h_item[nb][d] * norm[nb]
    {
      const int   nb = nbrs[(u0 + usr) * N_ + L];
      const float s  = nrm_src[nb];
      const float4* src = (const float4*)(h_item + (size_t)nb * D_);
      float4* dst = (float4*)&mail[L * MS];
#pragma unroll
      for (int k = 0; k < D_ / 4; ++k) {
        float4 v = src[k];
        v.x *= s; v.y *= s; v.z *= s; v.w *= s;
        dst[k] = v;
      }
    }
    __syncthreads();

    int sels[KS_];
    gram_greedy(mail, smat, L, hl, hs, sels);

    // pack selected rows into the shared 16-row tile (8 rows per user)
#pragma unroll
    for (int j = 0; j < KS_; ++j)
      ((float4*)&sel[(usr * KS_ + j) * MS])[L] =
          ((const float4*)&mail[sels[j] * MS])[L];
    __syncthreads();  // sel written before mail is overwritten next iteration
  }

  // ---- 6. rst = sum(selected) * N^-0.5 for both users ----
#pragma unroll
  for (int usr = 0; usr < 2; ++usr) {
    float4 r = {0.f, 0.f, 0.f, 0.f};
#pragma unroll
    for (int j = 0; j < KS_; ++j) {
      const float4 s4 = ((const float4*)&sel[(usr * KS_ + j) * MS])[L];
      r.x += s4.x; r.y += s4.y; r.z += s4.z; r.w += s4.w;
    }
    const float sc = 0.17677669529663687f;  // 32^-0.5
    r.x *= sc; r.y *= sc; r.z *= sc; r.w *= sc;
    ((float4*)(out + RST_BASE + (size_t)(u0 + usr) * D_))[L] = r;
  }

  // ---- 7. proj = tanh(S * W): M=16 (two users), N=128, K=128 ----
  for (int t = 0; t < 8; ++t) {
    const int n0 = 16 * t;
    v8f c = {};
    for (int k0 = 0; k0 < D_; k0 += 4) {
      const int kk = k0 + 2 * hs;
      const v2f a = *(const v2f*)&sel[hl * MS + kk];
      v2f b;
      b.x = Wp[(kk    ) * CCD_ + n0 + hl];   // W is K-major: two b32 loads
      b.y = Wp[(kk + 1) * CCD_ + n0 + hl];
      c = wmma_f32(a, b, c);
    }
#pragma unroll
    for (int v = 0; v < 8; ++v)
      pj[(v + 8 * hs) * MS + n0 + hl] = tanhf(c[v]);
  }
  __syncthreads();

  // ---- 8. logits = P * codes^T: M=16 (two users), N=32, K=128 ----
  {
    v8f l0 = {}, l1 = {};
    for (int k0 = 0; k0 < CCD_; k0 += 4) {
      const int kk = k0 + 2 * hs;
      const v2f a  = *(const v2f*)&pj[hl * MS + kk];
      const v2f b0 = *(const v2f*)&codes[( 0 + hl) * CCD_ + kk];  // contiguous
      const v2f b1 = *(const v2f*)&codes[(16 + hl) * CCD_ + kk];
      l0 = wmma_f32(a, b0, l0);
      l1 = wmma_f32(a, b1, l1);
    }
    // all 16 rows are real: packed k index = v + 8*hs (0-7 u0, 8-15 u1)
#pragma unroll
    for (int v = 0; v < 8; ++v) {
      lg[( 0 + hl) * LS + v + 8 * hs] = l0[v];
      lg[(16 + hl) * LS + v + 8 * hs] = l1[v];
    }
  }
  __syncthreads();

  // ---- 9. softmax over k per (user, code); lane L owns one code ----
#pragma unroll
  for (int usr = 0; usr < 2; ++usr) {
    const int kb = usr * KS_;
    float mx = -3.4e38f;
#pragma unroll
    for (int j = 0; j < KS_; ++j) mx = fmaxf(mx, lg[L * LS + kb + j]);
    float e[KS_]; float s = 0.0f;
#pragma unroll
    for (int j = 0; j < KS_; ++j) {
      e[j] = expf(lg[L * LS + kb + j] - mx);
      s += e[j];
    }
    const float inv = 1.0f / s;
#pragma unroll
    for (int j = 0; j < KS_; ++j) wt[L * LS + kb + j] = e[j] * inv;
  }
  __syncthreads();

  // ---- 10. muti_int = weights(32x8) * S(8x128), per user ----
#pragma unroll
  for (int usr = 0; usr < 2; ++usr) {
    const int kb = usr * KS_;
    float* ob = out + (size_t)(u0 + usr) * (KC_ * D_);
    for (int t = 0; t < 8; ++t) {
      const int n0 = 16 * t;
      v8f c0 = {}, c1 = {};
#pragma unroll
      for (int k0 = 0; k0 < KS_; k0 += 4) {
        const int kk = k0 + 2 * hs;
        const v2f a0 = *(const v2f*)&wt[( 0 + hl) * LS + kb + kk];
        const v2f a1 = *(const v2f*)&wt[(16 + hl) * LS + kb + kk];
        v2f b;
        b.x = sel[(kb + kk    ) * MS + n0 + hl];
        b.y = sel[(kb + kk + 1) * MS + n0 + hl];
        c0 = wmma_f32(a0, b, c0);
        c1 = wmma_f32(a1, b, c1);
      }
#pragma unroll
      for (int v = 0; v < 8; ++v) {
        ob[( 0 + v + 8 * hs) * D_ + n0 + hl] = c0[v];
        ob[(16 + v + 8 * hs) * D_ + n0 + hl] = c1[v];
      }
    }
  }
}

extern "C" void kernel_launch(void* const* d_in, const int* in_sizes, int n_in,
                              void* d_out, int out_size, void* d_ws, size_t ws_size,
                              hipStream_t stream) {
  (void)in_sizes; (void)n_in; (void)out_size; (void)ws_size;
  const float* h_item = (const float*)d_in[0];
  const int*   nbrs   = (const int*)d_in[1];
  const float* W      = (const float*)d_in[2];
  const float* codes  = (const float*)d_in[3];
  float* out = (float*)d_out;
  float* deg = (float*)d_ws;  // I_ floats of scratch

  deg_zero_kernel <<<(I_ + 255) / 256, 256, 0, stream>>>(deg);
  deg_count_kernel<<<(U_ * N_ + 255) / 256, 256, 0, stream>>>(nbrs, deg);
  deg_norm_kernel <<<(I_ + 255) / 256, 256, 0, stream>>>(deg);
  moerec_kernel   <<<U_ / 2, 32, 0, stream>>>(h_item, nbrs, W, codes, deg, out);
}